// AssociativeMemoryStep_67894843015277
// MI455X (gfx1250) — compile-verified
//
#include <hip/hip_runtime.h>
#include <hip/hip_bf16.h>
#include <cstdint>

// ---------------------------------------------------------------------------
// Fourier-parameterized linear-attention ("associative memory") forward pass.
// All GEMM work runs through V_WMMA_F32_16X16X4_F32 (fp32 WMMA, matches the
// fp32 reference numerics). The serial scan is re-expressed as chunked linear
// attention (chunk L=16): WMMA tile GEMMs with the 128x128 state M in LDS,
// one workgroup per batch, each wave owning 16 columns of M.  Chunk staging
// uses GLOBAL_LOAD_ASYNC_TO_LDS_B128 double-buffering (ASYNCcnt) to hide
// global->LDS latency on the serial critical path.
// ---------------------------------------------------------------------------

typedef __attribute__((ext_vector_type(2))) float v2f;
typedef __attribute__((ext_vector_type(8))) float v8f;

__device__ __forceinline__ v8f wmma4(v2f a, v2f b, v8f c) {
  // D = A(16x4) * B(4x16) + C(16x16), fp32
  return __builtin_amdgcn_wmma_f32_16x16x4_f32(
      /*neg_a=*/false, a, /*neg_b=*/false, b,
      /*c_mod=*/(short)0, c, /*reuse_a=*/false, /*reuse_b=*/false);
}

// A-fragment of a row-major matrix: element (m0+M, k0+K)
// lane L: M = L&15, K = 2*(L>>4) + g   (g = vgpr index 0/1)
__device__ __forceinline__ v2f fragA(const float* base, int stride, int m0,
                                     int k0, int lane) {
  int m = m0 + (lane & 15);
  int k = k0 + ((lane >> 4) << 1);
  v2f a;
  a.x = base[m * stride + k];
  a.y = base[m * stride + k + 1];
  return a;
}

// B-fragment of a row-major matrix: element (k0+K, n0+N)
// lane L: N = L&15, K = 2*(L>>4) + g
__device__ __forceinline__ v2f fragB(const float* base, int stride, int k0,
                                     int n0, int lane) {
  int n = n0 + (lane & 15);
  int k = k0 + ((lane >> 4) << 1);
  v2f b;
  b.x = base[k * stride + n];
  b.y = base[(k + 1) * stride + n];
  return b;
}

// Async global->LDS DMA, 16 bytes per lane, tracked by ASYNCcnt.
// VDST = per-lane LDS byte address (low 32 bits of generic LDS pointer).
__device__ __forceinline__ void async_b128(const float* g, float* l) {
  const uint32_t lds = (uint32_t)(uintptr_t)l;
  asm volatile("global_load_async_to_lds_b128 %0, %1, off"
               :
               : "v"(lds), "v"(g)
               : "memory");
}
__device__ __forceinline__ void wait_async_le6() {
  asm volatile("s_wait_asynccnt 0x6" ::: "memory");
}
__device__ __forceinline__ void wait_async_0() {
  asm volatile("s_wait_asynccnt 0x0" ::: "memory");
}

// ---------------------------------------------------------------------------
// Kernel 1: projection weights.  logits = basis @ coeffs.T  (V x C),
// softmax over V for q/k/v; raw logits for o.  One block per (column, which).
// ---------------------------------------------------------------------------
__global__ void fourier_weights_kernel(const float* __restrict__ basis,
                                       const float* __restrict__ qc,
                                       const float* __restrict__ kc,
                                       const float* __restrict__ vc,
                                       const float* __restrict__ oc,
                                       float* __restrict__ wq,
                                       float* __restrict__ wk,
                                       float* __restrict__ wv,
                                       float* __restrict__ wo) {
  __shared__ float cf[64];
  __shared__ float red[256];
  const int c = blockIdx.x;    // output channel 0..127
  const int w = blockIdx.y;    // 0=q 1=k 2=v 3=o
  const int tid = threadIdx.x; // 256 threads, 4 vocab rows each
  const float* coef = (w == 0) ? qc : (w == 1) ? kc : (w == 2) ? vc : oc;
  float* wout = (w == 0) ? wq : (w == 1) ? wk : (w == 2) ? wv : wo;

  if (tid < 64) cf[tid] = coef[c * 64 + tid];
  __syncthreads();

  float lg[4];
  for (int r = 0; r < 4; ++r) {
    const float* brow = basis + (size_t)(tid + r * 256) * 64;
    float s = 0.f;
    for (int f = 0; f < 64; ++f) s += brow[f] * cf[f];
    lg[r] = s;
  }
  if (w == 3) {  // o_w: no softmax
    for (int r = 0; r < 4; ++r) wout[(size_t)(tid + r * 256) * 128 + c] = lg[r];
    return;
  }
  // softmax over the 1024 vocab rows of this column
  float mx = fmaxf(fmaxf(lg[0], lg[1]), fmaxf(lg[2], lg[3]));
  red[tid] = mx;
  __syncthreads();
  for (int s = 128; s > 0; s >>= 1) {
    if (tid < s) red[tid] = fmaxf(red[tid], red[tid + s]);
    __syncthreads();
  }
  mx = red[0];
  __syncthreads();
  float e[4];
  float se = 0.f;
  for (int r = 0; r < 4; ++r) {
    e[r] = __expf(lg[r] - mx);
    se += e[r];
  }
  red[tid] = se;
  __syncthreads();
  for (int s = 128; s > 0; s >>= 1) {
    if (tid < s) red[tid] += red[tid + s];
    __syncthreads();
  }
  const float inv = 1.f / red[0];
  for (int r = 0; r < 4; ++r)
    wout[(size_t)(tid + r * 256) * 128 + c] = e[r] * inv;
}

// ---------------------------------------------------------------------------
// Kernel 2: fused q/k/v projection GEMM.  (16384 x 1024) @ (1024 x 128) x3.
// One block per 16-row M-tile; x tile staged in LDS (padded stride), each of
// 8 waves owns one 16-col N-tile and runs 3 accumulator chains sharing A.
// ---------------------------------------------------------------------------
#define XSTR 1028
__global__ void proj_qkv_kernel(const float* __restrict__ x,
                                const float* __restrict__ wq,
                                const float* __restrict__ wk,
                                const float* __restrict__ wv,
                                float* __restrict__ qo, float* __restrict__ ko,
                                float* __restrict__ vo) {
  extern __shared__ float xs[];  // 16 x XSTR floats
  const int tid = threadIdx.x;
  const int m0 = blockIdx.x * 16;

  // stage 16x1024 x-tile (4096 float4, 16 per thread, coalesced)
  for (int i = tid; i < 4096; i += 256) {
    const int row = i >> 8, c4 = (i & 255) << 2;
    const float4 val =
        *(const float4*)(x + (size_t)(m0 + row) * 1024 + c4);
    *(float4*)(xs + row * XSTR + c4) = val;
  }
  __syncthreads();

  const int wave = tid >> 5, lane = tid & 31;
  const int n0 = wave * 16;
  v8f aq = {}, ak = {}, av = {};
  for (int kk = 0; kk < 1024; kk += 4) {
    const v2f a = fragA(xs, XSTR, 0, kk, lane);
    const v2f bq = fragB(wq, 128, kk, n0, lane);
    const v2f bk = fragB(wk, 128, kk, n0, lane);
    const v2f bv = fragB(wv, 128, kk, n0, lane);
    aq = wmma4(a, bq, aq);
    ak = wmma4(a, bk, ak);
    av = wmma4(a, bv, av);
  }
  for (int g = 0; g < 8; ++g) {
    const int i = g + ((lane >> 4) << 3), j = lane & 15;
    const size_t off = (size_t)(m0 + i) * 128 + n0 + j;
    qo[off] = aq[g];
    ko[off] = ak[g];
    vo[off] = av[g];
  }
}

// ---------------------------------------------------------------------------
// Kernel 3: chunked linear-attention scan.  One block (8 waves) per batch.
// M (128x128) lives in LDS; wave w owns columns [16w,16w+16).
// Per chunk (L=16):  S' = mask_decay(Q K^T);  R = S'V + (d^i Q) M;
//                    M[:,w] = d^16 M[:,w] + (d^{15-j} K)^T V[:,w]
// Q/K/V chunk staging is double-buffered via GLOBAL_LOAD_ASYNC_TO_LDS_B128.
// ---------------------------------------------------------------------------
#define MSTR 132
#define CSTR 132
#define SSTR 17
#define CHB (3 * 16 * CSTR)  // one chunk buffer: Q,K,V each 16*CSTR

// Issue async DMA for one 16x128 chunk of q/k/v into an LDS buffer.
// 6 GLOBAL_LOAD_ASYNC_TO_LDS_B128 per thread (2 float4 rows x 3 matrices).
__device__ __forceinline__ void stage_chunk_async(const float* __restrict__ qb,
                                                  const float* __restrict__ kb,
                                                  const float* __restrict__ vb,
                                                  float* buf, int t0, int tid) {
  float* Qs = buf;
  float* Ks = buf + 16 * CSTR;
  float* Vs = buf + 2 * 16 * CSTR;
#pragma unroll
  for (int i = tid; i < 512; i += 256) {
    const int row = i >> 5, c4 = (i & 31) << 2;
    const size_t g = (size_t)(t0 + row) * 128 + c4;
    const int l = row * CSTR + c4;
    async_b128(qb + g, Qs + l);
    async_b128(kb + g, Ks + l);
    async_b128(vb + g, Vs + l);
  }
}

__global__ void scan_kernel(const float* __restrict__ q,
                            const float* __restrict__ k,
                            const float* __restrict__ v,
                            float* __restrict__ ret,
                            const float* __restrict__ decay_p) {
  extern __shared__ float smem[];
  float* Ml = smem;                    // 128*MSTR
  float* buf0 = Ml + 128 * MSTR;       // chunk buffer 0 (Q,K,V)
  float* buf1 = buf0 + CHB;            // chunk buffer 1 (Q,K,V)
  float* Ss = buf1 + CHB;              // 16*SSTR
  float* dp = Ss + 16 * SSTR;          // 17 decay powers d^0..d^16

  const int tid = threadIdx.x, lane = tid & 31, wave = tid >> 5;
  const int b = blockIdx.x;
  const float d = 1.f / (1.f + __expf(-decay_p[0]));  // sigmoid(decay)

  const size_t base = (size_t)b * 2048 * 128;
  const float* qb = q + base;
  const float* kb = k + base;
  const float* vb = v + base;
  float* rb = ret + base;

  // prefetch chunk 0 while we initialize M and the decay table
  stage_chunk_async(qb, kb, vb, buf0, 0, tid);

  for (int i = tid; i < 128 * MSTR; i += 256) Ml[i] = 0.f;  // M0 = 0
  if (tid == 0) {
    float p = 1.f;
    for (int i = 0; i <= 16; ++i) { dp[i] = p; p *= d; }
  }
  __syncthreads();

  const int n0 = wave * 16;
  const float qsc = dp[lane & 15];  // d^i for A-frag row i of Q
  const float d16 = dp[16];

  for (int chunk = 0; chunk < 128; ++chunk) {
    const int t0 = chunk * 16;
    float* cur = (chunk & 1) ? buf1 : buf0;
    float* nxt = (chunk & 1) ? buf0 : buf1;
    const float* Qs = cur;
    const float* Ks = cur + 16 * CSTR;
    const float* Vs = cur + 2 * 16 * CSTR;

    // overlap: launch next chunk's DMA, then wait only for current chunk
    // (async loads complete in order: <=6 outstanding -> current 6 are done)
    if (chunk + 1 < 128) {
      stage_chunk_async(qb, kb, vb, nxt, t0 + 16, tid);
      wait_async_le6();
    } else {
      wait_async_0();
    }
    __syncthreads();

    if (wave == 0) {  // S = Q K^T, then causal decay mask
      v8f s = {};
      for (int c0 = 0; c0 < 128; c0 += 4) {
        const v2f a = fragA(Qs, CSTR, 0, c0, lane);
        const v2f bt = fragA(Ks, CSTR, 0, c0, lane);  // B = K^T
        s = wmma4(a, bt, s);
      }
      for (int g = 0; g < 8; ++g) {
        const int i = g + ((lane >> 4) << 3), j = lane & 15;
        Ss[i * SSTR + j] = (j < i) ? s[g] * dp[i - 1 - j] : 0.f;
      }
    }
    __syncthreads();

    // retrieved tile (16 x 16) for this wave's columns
    v8f r = {};
    for (int kk = 0; kk < 16; kk += 4) {  // intra-chunk: S' @ V
      const v2f a = fragA(Ss, SSTR, 0, kk, lane);
      const v2f bb = fragB(Vs, CSTR, kk, n0, lane);
      r = wmma4(a, bb, r);
    }
    for (int kk = 0; kk < 128; kk += 4) {  // inter-chunk: (d^i Q) @ M
      v2f a = fragA(Qs, CSTR, 0, kk, lane);
      a.x *= qsc;
      a.y *= qsc;
      const v2f bb = fragB(Ml, MSTR, kk, n0, lane);
      r = wmma4(a, bb, r);
    }
    for (int g = 0; g < 8; ++g) {
      const int i = g + ((lane >> 4) << 3), j = lane & 15;
      rb[(size_t)(t0 + i) * 128 + n0 + j] = r[g];
    }

    // M[:, n0:n0+16] = d^16 * M + (decayed K)^T @ V   (wave-exclusive cols)
    const int jj = lane & 15, ib = (lane >> 4) << 3;
    for (int mt = 0; mt < 8; ++mt) {
      const int m0 = mt * 16;
      v8f acc;
      for (int g = 0; g < 8; ++g)
        acc[g] = Ml[(m0 + g + ib) * MSTR + n0 + jj] * d16;
      for (int kk = 0; kk < 16; kk += 4) {
        const int m = m0 + (lane & 15);
        const int kx = kk + ((lane >> 4) << 1);
        v2f a;  // A = (d^{15-j} K)^T : element (m, kx) = dp[15-kx]*K[kx][m]
        a.x = dp[15 - kx] * Ks[kx * CSTR + m];
        a.y = dp[14 - kx] * Ks[(kx + 1) * CSTR + m];
        const v2f bb = fragB(Vs, CSTR, kk, n0, lane);
        acc = wmma4(a, bb, acc);
      }
      for (int g = 0; g < 8; ++g)
        Ml[(m0 + g + ib) * MSTR + n0 + jj] = acc[g];
    }
    // all waves done reading `cur` before next iteration's DMA overwrites it
    __syncthreads();
  }
}

// ---------------------------------------------------------------------------
// Kernel 4: output GEMM  out = (retrieved @ o_w.T) * out_scale
// (16384 x 128) @ (128 x 1024).  Block = one 16-row tile x 128 v-columns.
// ---------------------------------------------------------------------------
#define RSTR 132
__global__ void out_proj_kernel(const float* __restrict__ r,
                                const float* __restrict__ wo,
                                const float* __restrict__ osp,
                                float* __restrict__ out) {
  __shared__ float rs[16 * RSTR];
  const int tid = threadIdx.x;
  const int m0 = blockIdx.x * 16;
  for (int i = tid; i < 512; i += 256) {  // stage 16x128 retrieved tile
    const int row = i >> 5, c4 = (i & 31) << 2;
    *(float4*)(rs + row * RSTR + c4) =
        *(const float4*)(r + (size_t)(m0 + row) * 128 + c4);
  }
  __syncthreads();
  const int wave = tid >> 5, lane = tid & 31;
  const int n0 = blockIdx.y * 128 + wave * 16;  // v-column tile
  v8f acc = {};
  for (int kk = 0; kk < 128; kk += 4) {
    const v2f a = fragA(rs, RSTR, 0, kk, lane);
    // B[k=c, n=v] = o_w[v][c] -> transposed read == fragA addressing on wo
    const v2f bb = fragA(wo, 128, n0, kk, lane);
    acc = wmma4(a, bb, acc);
  }
  const float os = *osp;
  for (int g = 0; g < 8; ++g) {
    const int i = g + ((lane >> 4) << 3), j = lane & 15;
    out[(size_t)(m0 + i) * 1024 + n0 + j] = acc[g] * os;
  }
}

// ---------------------------------------------------------------------------
extern "C" void kernel_launch(void* const* d_in, const int* in_sizes, int n_in,
                              void* d_out, int out_size, void* d_ws,
                              size_t ws_size, hipStream_t stream) {
  (void)in_sizes; (void)n_in; (void)out_size; (void)ws_size;
  const float* x      = (const float*)d_in[0];
  const float* basis  = (const float*)d_in[1];
  const float* qc     = (const float*)d_in[2];
  const float* kc     = (const float*)d_in[3];
  const float* vc     = (const float*)d_in[4];
  const float* oc     = (const float*)d_in[5];
  const float* decay  = (const float*)d_in[6];
  const float* oscale = (const float*)d_in[7];
  float* out = (float*)d_out;

  // workspace layout (floats)
  float* ws = (float*)d_ws;
  const size_t WSZ = 1024 * 128;        // 131072 per weight matrix
  const size_t PSZ = 8 * 2048 * 128;    // 2097152 per (B,T,C) tensor
  float* wq = ws;
  float* wk = wq + WSZ;
  float* wv = wk + WSZ;
  float* wo = wv + WSZ;
  float* q  = wo + WSZ;
  float* k  = q + PSZ;
  float* v  = k + PSZ;
  float* r  = v + PSZ;

  dim3 g1(128, 4);
  fourier_weights_kernel<<<g1, 256, 0, stream>>>(basis, qc, kc, vc, oc,
                                                 wq, wk, wv, wo);

  const size_t proj_lds = (size_t)16 * XSTR * sizeof(float);  // 65792 B
  proj_qkv_kernel<<<1024, 256, proj_lds, stream>>>(x, wq, wk, wv, q, k, v);

  const size_t scan_lds =
      (size_t)(128 * MSTR + 2 * CHB + 16 * SSTR + 17) * sizeof(float);
  scan_kernel<<<8, 256, scan_lds, stream>>>(q, k, v, r, decay);

  dim3 g4(1024, 8);
  out_proj_kernel<<<g4, 256, 0, stream>>>(r, wo, oscale, out);
}